// Treatformer_32375463477458
// MI455X (gfx1250) — compile-verified
//
#include <hip/hip_runtime.h>
#include <hip/hip_bf16.h>

// Model dims
constexpr int TT  = 512;   // sequence length
constexpr int NB  = 32;    // batch
constexpr int DD  = 512;   // model dim
constexpr int HH  = 8;     // heads
constexpr int DHD = 64;    // head dim
constexpr int FFH = 2048;  // ff hidden
constexpr int MHD = 50;    // mlp-head hidden
constexpr int MROWS = TT * NB; // 16384

typedef float v2f __attribute__((ext_vector_type(2)));
typedef float v4f __attribute__((ext_vector_type(4)));
typedef float v8f __attribute__((ext_vector_type(8)));
typedef int   v4i __attribute__((ext_vector_type(4)));

// ---- CDNA5 async global->LDS staging (guarded; falls back to VGPR staging) --
#if defined(__has_builtin)
# if __has_builtin(__builtin_amdgcn_global_load_async_to_lds_b128)
#  define ASYNC_LDS 1
# endif
#endif
#ifndef ASYNC_LDS
# define ASYNC_LDS 0
#endif

__device__ __forceinline__ void cp16_g2l(const float* gp, float* lp) {
#if ASYNC_LDS
  __builtin_amdgcn_global_load_async_to_lds_b128(
      (__attribute__((address_space(1))) v4i*)gp,
      (__attribute__((address_space(3))) v4i*)lp, 0, 0);
#else
  *(v4f*)lp = *(const v4f*)gp;
#endif
}

__device__ __forceinline__ void wait_async0() {
#if ASYNC_LDS
# if defined(__has_builtin) && __has_builtin(__builtin_amdgcn_s_wait_asynccnt)
  __builtin_amdgcn_s_wait_asynccnt(0);
# else
  asm volatile("s_wait_asynccnt 0" ::: "memory");
# endif
#endif
}

__device__ __forceinline__ v8f wmma_f32(v2f a, v2f b, v8f c) {
  // D = A(16x4 f32) * B(4x16 f32) + C(16x16 f32)
  return __builtin_amdgcn_wmma_f32_16x16x4_f32(false, a, false, b, (short)0, c,
                                               false, false);
}

// ---------------------------------------------------------------------------
// 1) e-tables: e[k][delta][d], delta = i-j+511 in [0,1022]
// ---------------------------------------------------------------------------
__global__ __launch_bounds__(256) void etab_kernel(
    const float* __restrict__ w1a, const float* __restrict__ b1a,
    const float* __restrict__ w2a, const float* __restrict__ b2a,
    const float* __restrict__ w1b, const float* __restrict__ b1b,
    const float* __restrict__ w2b, const float* __restrict__ b2b,
    float* __restrict__ etab) {
  int idx = blockIdx.x;                 // 0..1022
  float td = (float)(idx - (TT - 1));   // i - j
  int tid = threadIdx.x;
  __shared__ float h1[64], h2[64];
  if (tid < MHD) {
    h1[tid] = fmaxf(td * w1a[tid] + b1a[tid], 0.f);
    h2[tid] = fmaxf(td * w1b[tid] + b1b[tid], 0.f);
  }
  __syncthreads();
  for (int d = tid; d < DD; d += 256) {
    float s1 = b2a[d], s2 = b2b[d];
#pragma unroll 5
    for (int m = 0; m < MHD; ++m) {
      s1 += h1[m] * w2a[d * MHD + m];
      s2 += h2[m] * w2b[d * MHD + m];
    }
    etab[(size_t)idx * DD + d] = s1;
    etab[(size_t)(1023 + idx) * DD + d] = s2;
  }
}

// ---------------------------------------------------------------------------
// 2) Generic WMMA f32 GEMM: C[M,Nc] = act(A[M,K] @ W[Nc,K]^T + bias)
//    Block tile 64x128, 8 waves each 32x32 (2x2 WMMA tiles).
//    K staged by 16 through double-buffered LDS; async DMA when available;
//    one barrier per stage (issue(s+1) after data-ready barrier of s).
// ---------------------------------------------------------------------------
constexpr int LDA_S = 20;  // padded LDS row stride (conflict-free, 16B aligned)

__global__ __launch_bounds__(256) void gemm_wmma(
    const float* __restrict__ A, const float* __restrict__ W,
    const float* __restrict__ bias, float* __restrict__ C,
    int M, int Nc, int K, int relu) {
  __shared__ float As[2][64 * LDA_S];
  __shared__ float Ws[2][128 * LDA_S];

  int tid = threadIdx.x;
  int lane = tid & 31;
  int w = tid >> 5;
  int wr = w & 1;        // 0..1 -> 32-row strip
  int wc = w >> 1;       // 0..3 -> 32-col strip
  int lo = lane & 15;
  int kh = lane >> 4;

  size_t bm = (size_t)blockIdx.y * 64;
  size_t bn = (size_t)blockIdx.x * 128;

  int ra = tid >> 2, sega = (tid & 3) * 4;

  auto stage = [&](int buf, int ks) {
    // A tile 64x16 (256 b128 transfers, 1/thread)
    cp16_g2l(&A[(bm + ra) * (size_t)K + ks + sega], &As[buf][ra * LDA_S + sega]);
    // W tile 128x16 (512 b128 transfers, 2/thread)
#pragma unroll
    for (int rep = 0; rep < 2; ++rep) {
      int id2 = tid + rep * 256;
      int rw = id2 >> 2, sg = (id2 & 3) * 4;
      cp16_g2l(&W[(bn + rw) * (size_t)K + ks + sg], &Ws[buf][rw * LDA_S + sg]);
    }
  };

  v8f acc[2][2] = {};
  int nst = K / 16;
  stage(0, 0);
  for (int s = 0; s < nst; ++s) {
    int cur = s & 1;
    wait_async0();        // this wave's DMA for buffer `cur` has landed
    __syncthreads();      // everyone's has landed (also drains DScnt)
    if (s + 1 < nst) stage(cur ^ 1, (s + 1) * 16);
    const float* Ab = As[cur];
    const float* Wb = Ws[cur];
#pragma unroll
    for (int kk = 0; kk < 16; kk += 4) {
      v2f a0 = *(const v2f*)&Ab[(wr * 32 + lo) * LDA_S + kk + 2 * kh];
      v2f a1 = *(const v2f*)&Ab[(wr * 32 + 16 + lo) * LDA_S + kk + 2 * kh];
      v2f b0 = *(const v2f*)&Wb[(wc * 32 + lo) * LDA_S + kk + 2 * kh];
      v2f b1 = *(const v2f*)&Wb[(wc * 32 + 16 + lo) * LDA_S + kk + 2 * kh];
      acc[0][0] = wmma_f32(a0, b0, acc[0][0]);
      acc[0][1] = wmma_f32(a0, b1, acc[0][1]);
      acc[1][0] = wmma_f32(a1, b0, acc[1][0]);
      acc[1][1] = wmma_f32(a1, b1, acc[1][1]);
    }
  }

#pragma unroll
  for (int ti = 0; ti < 2; ++ti) {
#pragma unroll
    for (int tj = 0; tj < 2; ++tj) {
      int col = (int)bn + wc * 32 + tj * 16 + lo;
      float bv = bias ? bias[col] : 0.f;
#pragma unroll
      for (int e = 0; e < 8; ++e) {
        int row = (int)bm + wr * 32 + ti * 16 + e + 8 * kh;
        float val = acc[ti][tj][e] + bv;
        if (relu) val = fmaxf(val, 0.f);
        C[(size_t)row * Nc + col] = val;
      }
    }
  }
}

// ---------------------------------------------------------------------------
// 3) Attention: block = (n, h, 32-query tile). WMMA for S=QK^T and O=PV.
// ---------------------------------------------------------------------------
__global__ __launch_bounds__(256) void attn_kernel(
    const float* __restrict__ qkv, const float* __restrict__ src_mask,
    float* __restrict__ obuf) {
  __shared__ float Qs[32 * 68];
  __shared__ float Ks[64 * 68];
  __shared__ float VsT[64 * 68];   // [dh][key]
  __shared__ float Ps[32 * 516];   // padded scores/probs

  int bid = blockIdx.x;
  int qt = bid & 15;          // query tile (32 rows)
  int h = (bid >> 4) & 7;
  int n = bid >> 7;

  int tid = threadIdx.x;
  int lane = tid & 31;
  int w = tid >> 5;
  int wr = w & 1;             // 16-row tile within 32 queries
  int wc = w >> 1;            // 16-col tile within 64 keys / 64 dh
  int lo = lane & 15;
  int kh = lane >> 4;

  const size_t rowStride = (size_t)NB * (3 * DD);   // qkv row stride
  const float* qbase = qkv + (size_t)(qt * 32) * rowStride + (size_t)n * (3 * DD) + h * DHD;

  // Load Q tile: 32 rows x 16 float4
#pragma unroll
  for (int rep = 0; rep < 2; ++rep) {
    int idx = tid + rep * 256;
    int q = idx >> 4, seg = idx & 15;
    v4f t = *(const v4f*)(qbase + (size_t)q * rowStride + seg * 4);
    *(v4f*)&Qs[q * 68 + seg * 4] = t;
  }
  __syncthreads();

  // ---- scores S = Q K^T * (1/8) ----
  for (int kt = 0; kt < 8; ++kt) {
    const float* kbase = qkv + (size_t)(kt * 64) * rowStride + (size_t)n * (3 * DD) + DD + h * DHD;
#pragma unroll
    for (int rep = 0; rep < 4; ++rep) {
      int idx = tid + rep * 256;
      int kr = idx >> 4, seg = idx & 15;
      v4f t = *(const v4f*)(kbase + (size_t)kr * rowStride + seg * 4);
      *(v4f*)&Ks[kr * 68 + seg * 4] = t;
    }
    __syncthreads();
    v8f sacc = {};
#pragma unroll
    for (int kk = 0; kk < DHD; kk += 4) {
      v2f a = *(const v2f*)&Qs[(wr * 16 + lo) * 68 + kk + 2 * kh];
      v2f b = *(const v2f*)&Ks[(wc * 16 + lo) * 68 + kk + 2 * kh];
      sacc = wmma_f32(a, b, sacc);
    }
#pragma unroll
    for (int e = 0; e < 8; ++e)
      Ps[(wr * 16 + e + 8 * kh) * 516 + kt * 64 + wc * 16 + lo] = sacc[e] * 0.125f;
    __syncthreads();
  }

  // ---- softmax (additive mask; torch tile(H,1,1).reshape(N,H) => (n*H+h)%N) ----
  if (tid < 32) {
    int q = tid;
    int qg = qt * 32 + q;
    int mi = (n * HH + h) & (NB - 1);
    const float* mrow = src_mask + ((size_t)mi * TT + qg) * TT;
    float mx = -3.4e38f;
    for (int j = 0; j < TT; ++j) {
      float p = Ps[q * 516 + j] + mrow[j];
      Ps[q * 516 + j] = p;
      mx = fmaxf(mx, p);
    }
    float s = 0.f;
    for (int j = 0; j < TT; ++j) {
      float e = __expf(Ps[q * 516 + j] - mx);
      Ps[q * 516 + j] = e;
      s += e;
    }
    float inv = 1.f / s;
    for (int j = 0; j < TT; ++j) Ps[q * 516 + j] *= inv;
  }
  __syncthreads();

  // ---- O = P V ----
  v8f oacc = {};
  for (int kt = 0; kt < 8; ++kt) {
    const float* vbase = qkv + (size_t)(kt * 64) * rowStride + (size_t)n * (3 * DD) + 2 * DD + h * DHD;
#pragma unroll
    for (int rep = 0; rep < 4; ++rep) {
      int idx = tid + rep * 256;
      int kr = idx >> 4, seg = idx & 15;
      v4f t = *(const v4f*)(vbase + (size_t)kr * rowStride + seg * 4);
      VsT[(seg * 4 + 0) * 68 + kr] = t.x;
      VsT[(seg * 4 + 1) * 68 + kr] = t.y;
      VsT[(seg * 4 + 2) * 68 + kr] = t.z;
      VsT[(seg * 4 + 3) * 68 + kr] = t.w;
    }
    __syncthreads();
#pragma unroll
    for (int kk = 0; kk < 64; kk += 4) {
      v2f a = *(const v2f*)&Ps[(wr * 16 + lo) * 516 + kt * 64 + kk + 2 * kh];
      v2f b = *(const v2f*)&VsT[(wc * 16 + lo) * 68 + kk + 2 * kh];
      oacc = wmma_f32(a, b, oacc);
    }
    __syncthreads();
  }
  // store O in [T,N,D] layout (undoes the head transpose)
  {
    int dh = wc * 16 + lo;
#pragma unroll
    for (int e = 0; e < 8; ++e) {
      int q = qt * 32 + wr * 16 + e + 8 * kh;
      obuf[((size_t)q * NB + n) * DD + h * DHD + dh] = oacc[e];
    }
  }
}

// ---------------------------------------------------------------------------
// 4) treat additive effect: src2[i,n,:] += t1*sum_j(!m)*e1[i-j] + t2*...*e2
// ---------------------------------------------------------------------------
__global__ __launch_bounds__(256) void treat_add_kernel(
    const float* __restrict__ etab, const unsigned char* __restrict__ maskb,
    const float* __restrict__ src_treat, float* __restrict__ src2) {
  int i = blockIdx.x;   // query position
  int n = blockIdx.y;   // batch
  int tid = threadIdx.x;
  __shared__ unsigned char mrow[TT];
  if (tid < TT / 4)
    ((unsigned int*)mrow)[tid] =
        ((const unsigned int*)(maskb + ((size_t)n * TT + i) * TT))[tid];
  __syncthreads();

  const float* e1 = etab;
  const float* e2 = etab + (size_t)1023 * DD;
  int d0 = tid, d1 = tid + 256;
  float s1a = 0.f, s1b = 0.f, s2a = 0.f, s2b = 0.f;
  for (int j = 0; j < TT; ++j) {
    if (!mrow[j]) {  // mask = ~src_mask_treat
      size_t idx = (size_t)(i - j + (TT - 1)) * DD;
      s1a += e1[idx + d0];
      s1b += e1[idx + d1];
      s2a += e2[idx + d0];
      s2b += e2[idx + d1];
    }
  }
  float t1 = src_treat[((size_t)i * NB + n) * 2 + 0];
  float t2 = src_treat[((size_t)i * NB + n) * 2 + 1];
  size_t r = ((size_t)i * NB + n) * DD;
  src2[r + d0] += t1 * s1a + t2 * s2a;
  src2[r + d1] += t1 * s1b + t2 * s2b;
}

// ---------------------------------------------------------------------------
// 5) LayerNorm(a + b) * g + beta, one row (512) per block
// ---------------------------------------------------------------------------
__global__ __launch_bounds__(256) void ln_kernel(
    const float* __restrict__ a, const float* __restrict__ b,
    const float* __restrict__ g, const float* __restrict__ beta,
    float* __restrict__ out) {
  int r = blockIdx.x;
  int tid = threadIdx.x;
  __shared__ float red[256];
  size_t base = (size_t)r * DD;
  float x0 = a[base + tid] + b[base + tid];
  float x1 = a[base + tid + 256] + b[base + tid + 256];
  red[tid] = x0 + x1;
  __syncthreads();
  for (int s = 128; s > 0; s >>= 1) {
    if (tid < s) red[tid] += red[tid + s];
    __syncthreads();
  }
  float mean = red[0] * (1.f / DD);
  __syncthreads();
  float dv0 = x0 - mean, dv1 = x1 - mean;
  red[tid] = dv0 * dv0 + dv1 * dv1;
  __syncthreads();
  for (int s = 128; s > 0; s >>= 1) {
    if (tid < s) red[tid] += red[tid + s];
    __syncthreads();
  }
  float inv = rsqrtf(red[0] * (1.f / DD) + 1e-5f);
  out[base + tid] = dv0 * inv * g[tid] + beta[tid];
  out[base + tid + 256] = dv1 * inv * g[tid + 256] + beta[tid + 256];
}

// ---------------------------------------------------------------------------
// Host orchestration
// ---------------------------------------------------------------------------
extern "C" void kernel_launch(void* const* d_in, const int* in_sizes, int n_in,
                              void* d_out, int out_size, void* d_ws, size_t ws_size,
                              hipStream_t stream) {
  (void)in_sizes; (void)n_in; (void)out_size; (void)ws_size;
  const float* src        = (const float*)d_in[0];
  const float* src_treat  = (const float*)d_in[1];
  const float* src_mask   = (const float*)d_in[2];
  const unsigned char* src_mask_treat = (const unsigned char*)d_in[3];
  const float* W_map = (const float*)d_in[4];
  const float* b_map = (const float*)d_in[5];
  const float* Wqkv  = (const float*)d_in[6];
  const float* bqkv  = (const float*)d_in[7];
  const float* Wo    = (const float*)d_in[8];
  const float* bo    = (const float*)d_in[9];
  const float* m1w1  = (const float*)d_in[10];
  const float* m1b1  = (const float*)d_in[11];
  const float* m1w2  = (const float*)d_in[12];
  const float* m1b2  = (const float*)d_in[13];
  const float* m2w1  = (const float*)d_in[14];
  const float* m2b1  = (const float*)d_in[15];
  const float* m2w2  = (const float*)d_in[16];
  const float* m2b2  = (const float*)d_in[17];
  const float* W1    = (const float*)d_in[18];
  const float* b1    = (const float*)d_in[19];
  const float* W2    = (const float*)d_in[20];
  const float* b2    = (const float*)d_in[21];
  const float* ln1g  = (const float*)d_in[22];
  const float* ln1b  = (const float*)d_in[23];
  const float* ln2g  = (const float*)d_in[24];
  const float* ln2b  = (const float*)d_in[25];
  float* out = (float*)d_out;

  // workspace carve-up (floats)
  float* ws = (float*)d_ws;
  const size_t ROW_SZ  = (size_t)MROWS * DD;             // 8.39M
  const size_t QKV_SZ  = (size_t)MROWS * 3 * DD;         // 25.2M
  float* etab = ws;                 size_t off = 1 << 20; // 1M-float pad >= etab
  float* xbuf = ws + off;           off += ROW_SZ;        // x, later reused as ff2
  float* qkv  = ws + off;           off += QKV_SZ;
  float* obuf = ws + off;           off += ROW_SZ;
  float* src2 = ws + off;           off += ROW_SZ;
  float* hbuf = ws + off;           off += ROW_SZ;
  float* ff1  = ws + off;           off += (size_t)MROWS * FFH;
  float* ff2  = xbuf;               // alias: x is dead after qkv GEMM

  dim3 blk(256);

  // 1) e-tables
  etab_kernel<<<dim3(2 * TT - 1), blk, 0, stream>>>(
      m1w1, m1b1, m1w2, m1b2, m2w1, m2b1, m2w2, m2b2, etab);

  // 2) x = src @ W_map^T + b_map
  gemm_wmma<<<dim3(DD / 128, MROWS / 64), blk, 0, stream>>>(
      src, W_map, b_map, xbuf, MROWS, DD, DD, 0);

  // 3) qkv = x @ Wqkv^T + bqkv
  gemm_wmma<<<dim3(3 * DD / 128, MROWS / 64), blk, 0, stream>>>(
      xbuf, Wqkv, bqkv, qkv, MROWS, 3 * DD, DD, 0);

  // 4) attention -> obuf [T,N,D]
  attn_kernel<<<dim3(NB * HH * (TT / 32)), blk, 0, stream>>>(qkv, src_mask, obuf);

  // 5) src2 = o @ Wo^T + bo
  gemm_wmma<<<dim3(DD / 128, MROWS / 64), blk, 0, stream>>>(
      obuf, Wo, bo, src2, MROWS, DD, DD, 0);

  // 6) src2 += treat additive effect
  treat_add_kernel<<<dim3(TT, NB), blk, 0, stream>>>(etab, src_mask_treat,
                                                     src_treat, src2);

  // 7) h = LN(src + src2)
  ln_kernel<<<dim3(MROWS), blk, 0, stream>>>(src, src2, ln1g, ln1b, hbuf);

  // 8) ff1 = relu(h @ W1^T + b1)
  gemm_wmma<<<dim3(FFH / 128, MROWS / 64), blk, 0, stream>>>(
      hbuf, W1, b1, ff1, MROWS, FFH, DD, 1);

  // 9) ff2 = ff1 @ W2^T + b2
  gemm_wmma<<<dim3(DD / 128, MROWS / 64), blk, 0, stream>>>(
      ff1, W2, b2, ff2, MROWS, DD, FFH, 0);

  // 10) out = LN(h + ff2)
  ln_kernel<<<dim3(MROWS), blk, 0, stream>>>(hbuf, ff2, ln2g, ln2b, out);
}